// Attention_85332410236931
// MI455X (gfx1250) — compile-verified
//
#include <hip/hip_runtime.h>
#include <hip/hip_bf16.h>

// ---------------------------------------------------------------------------
// Problem: N=64, L=2048, D=512, A=128, S=512
// Algebraic rewrite (exact):
//   kq(n,:)   = Wk @ (Wq^T s_n + bq)            -> two tiny WMMA GEMMs
//   e(n,l)    = H(n,l,:) . kq(n,:)   (+ const per n: cancels in softmax)
//   w(n,l)    = softmax over l < len(n) only  (== masked softmax + L1 renorm)
//   context   = (sum_l w * H(n,l,:)) @ Wv + bv  -> one tiny WMMA GEMM
// Single HBM pass over the 256MB H tensor via online softmax.
// ---------------------------------------------------------------------------

typedef float v2f __attribute__((ext_vector_type(2)));
typedef float v8f __attribute__((ext_vector_type(8)));

#define N_B   64
#define L_B   2048
#define D_B   512
#define A_B   128
#define SPLIT 8

// ---------------------------------------------------------------------------
// Generic small f32 GEMM: D = A(MxK) @ B(KxN) (+bias), exact f32 via
// V_WMMA_F32_16X16X4_F32. One wave per 16x16 output tile.
// BT=false: B element [k][n] = B[k*ldb + n]; BT=true: = B[n*ldb + k].
// ---------------------------------------------------------------------------
template<bool BT>
__global__ void wmma_gemm_f32(const float* __restrict__ Am,
                              const float* __restrict__ Bm,
                              const float* __restrict__ bias,
                              float* __restrict__ Dm,
                              int K, int lda, int ldb, int ldd, int ntiles_n) {
    const int tile = blockIdx.x;
    const int mt = tile / ntiles_n;
    const int nt = tile % ntiles_n;
    const int lane = threadIdx.x & 31;
    const int hh   = lane >> 4;      // which half of the wave (0/1)
    const int lm   = lane & 15;
    const int row  = mt * 16 + lm;   // A-matrix row owned by this lane
    const int col  = nt * 16 + lm;   // B/D column owned by this lane

    v8f c = {};
    for (int k0 = 0; k0 < K; k0 += 4) {
        const int ka = k0 + 2 * hh;  // lanes 0-15: k0,k0+1 ; lanes 16-31: k0+2,k0+3
        v2f a, b;
        a.x = Am[row * lda + ka];
        a.y = Am[row * lda + ka + 1];
        if (BT) {
            b.x = Bm[col * ldb + ka];
            b.y = Bm[col * ldb + ka + 1];
        } else {
            b.x = Bm[(ka)     * ldb + col];
            b.y = Bm[(ka + 1) * ldb + col];
        }
        c = __builtin_amdgcn_wmma_f32_16x16x4_f32(false, a, false, b,
                                                  (short)0, c, false, false);
    }
    const float bb = bias ? bias[col] : 0.0f;
#pragma unroll
    for (int j = 0; j < 8; ++j) {
        const int m = mt * 16 + j + 8 * hh;   // VGPR j: M=j (lo half) / M=8+j (hi half)
        Dm[m * ldd + col] = c[j] + bb;
    }
}

// ---------------------------------------------------------------------------
// Fused streaming pass over H: per (n, split) workgroup of 8 waves.
// Each wave owns 32 consecutive l positions; per l it keeps the whole 512-wide
// row in registers (16 floats/lane), computes the energy dot against cached
// kq, does an online-softmax update, and accumulates p*h into 16 register
// accumulators. Raw energies are parked in the w output region.
// ---------------------------------------------------------------------------
__global__ void attn_stream(const float* __restrict__ H,
                            const float* __restrict__ kq,
                            const int*   __restrict__ lens,
                            float* __restrict__ e_out,      // (N,L) raw energies
                            float* __restrict__ part_hbar,  // (N*SPLIT, D)
                            float* __restrict__ part_mz) {  // (N*SPLIT, 2)
    const int n    = blockIdx.x / SPLIT;
    const int s    = blockIdx.x % SPLIT;
    const int wave = threadIdx.x >> 5;
    const int lane = threadIdx.x & 31;
    const int len  = lens[n];

    // kq ownership: lane covers d = 4*lane + 128*j, j=0..3 (coalesced float4)
    const float4* kqv = (const float4*)(kq + (size_t)n * D_B);
    float4 kq4[4];
#pragma unroll
    for (int j = 0; j < 4; ++j) kq4[j] = kqv[lane + 32 * j];

    float4 acc[4] = {};
    float  m = -1e30f, Z = 0.0f;

    const int l0 = s * (L_B / SPLIT) + wave * 32;
    const float* Hn = H + (size_t)n * L_B * D_B;
    int cnt = len - l0; if (cnt > 32) cnt = 32; if (cnt < 0) cnt = 0;

    for (int i = 0; i < cnt; ++i) {
        const int l = l0 + i;
        const float4* h4 = (const float4*)(Hn + (size_t)l * D_B);
        if (i + 1 < cnt)
            __builtin_prefetch(Hn + (size_t)(l + 1) * D_B + 4 * lane, 0, 1);
        float4 hv[4];
#pragma unroll
        for (int j = 0; j < 4; ++j) hv[j] = h4[lane + 32 * j];

        float e = 0.0f;
#pragma unroll
        for (int j = 0; j < 4; ++j)
            e += hv[j].x * kq4[j].x + hv[j].y * kq4[j].y +
                 hv[j].z * kq4[j].z + hv[j].w * kq4[j].w;
#pragma unroll
        for (int off = 16; off; off >>= 1) e += __shfl_xor(e, off, 32);

        if (lane == 0) e_out[(size_t)n * L_B + l] = e;

        const float mn    = fmaxf(m, e);
        const float scale = __expf(m - mn);   // 1 when m unchanged, 0 on first iter
        const float p     = __expf(e - mn);
        Z = Z * scale + p;
#pragma unroll
        for (int j = 0; j < 4; ++j) {
            acc[j].x = acc[j].x * scale + p * hv[j].x;
            acc[j].y = acc[j].y * scale + p * hv[j].y;
            acc[j].z = acc[j].z * scale + p * hv[j].z;
            acc[j].w = acc[j].w * scale + p * hv[j].w;
        }
        m = mn;
    }

    // Combine the 8 waves of this workgroup in LDS.
    __shared__ float sm[8], sZ[8];
    __shared__ float shb[8 * D_B];            // 16 KB
    if (lane == 0) { sm[wave] = m; sZ[wave] = Z; }
#pragma unroll
    for (int j = 0; j < 4; ++j)
        ((float4*)(shb + wave * D_B))[lane + 32 * j] = acc[j];
    __syncthreads();

    float mg = -1e30f;
#pragma unroll
    for (int wv = 0; wv < 8; ++wv) mg = fmaxf(mg, sm[wv]);
    float zg = 0.0f, h0 = 0.0f, h1 = 0.0f;
    const int d0 = 2 * threadIdx.x;
#pragma unroll
    for (int wv = 0; wv < 8; ++wv) {
        const float sc = __expf(sm[wv] - mg);
        zg += sZ[wv] * sc;
        h0 += shb[wv * D_B + d0]     * sc;
        h1 += shb[wv * D_B + d0 + 1] * sc;
    }
    const int pidx = blockIdx.x;               // n*SPLIT + s
    part_hbar[(size_t)pidx * D_B + d0]     = h0;
    part_hbar[(size_t)pidx * D_B + d0 + 1] = h1;
    if (threadIdx.x == 0) { part_mz[2 * pidx] = mg; part_mz[2 * pidx + 1] = zg; }
}

// ---------------------------------------------------------------------------
// Combine SPLIT partials per batch row -> normalized hbar + final (m, Z).
// ---------------------------------------------------------------------------
__global__ void combine_splits(const float* __restrict__ part_hbar,
                               const float* __restrict__ part_mz,
                               float* __restrict__ hbar_norm,   // (N, D)
                               float* __restrict__ mz_final) {  // (N, 2)
    const int n = blockIdx.x;
    float mg = -1e30f;
#pragma unroll
    for (int s = 0; s < SPLIT; ++s) mg = fmaxf(mg, part_mz[2 * (n * SPLIT + s)]);
    float zg = 0.0f, h0 = 0.0f, h1 = 0.0f;
    const int d0 = 2 * threadIdx.x;
#pragma unroll
    for (int s = 0; s < SPLIT; ++s) {
        const int p = n * SPLIT + s;
        const float sc = __expf(part_mz[2 * p] - mg);
        zg += part_mz[2 * p + 1] * sc;
        h0 += part_hbar[(size_t)p * D_B + d0]     * sc;
        h1 += part_hbar[(size_t)p * D_B + d0 + 1] * sc;
    }
    const float inv = 1.0f / fmaxf(zg, 1e-12f);
    hbar_norm[(size_t)n * D_B + d0]     = h0 * inv;
    hbar_norm[(size_t)n * D_B + d0 + 1] = h1 * inv;
    if (threadIdx.x == 0) { mz_final[2 * n] = mg; mz_final[2 * n + 1] = zg; }
}

// ---------------------------------------------------------------------------
// Rewrite raw energies (in the w output region) as masked softmax weights.
// ---------------------------------------------------------------------------
__global__ void finalize_w(float* __restrict__ w,
                           const float* __restrict__ mz_final,
                           const int* __restrict__ lens) {
    const int idx = blockIdx.x * blockDim.x + threadIdx.x;
    const int n = idx >> 11;           // /2048
    const int l = idx & (L_B - 1);
    const float m = mz_final[2 * n];
    const float Z = fmaxf(mz_final[2 * n + 1], 1e-30f);
    float val = 0.0f;
    if (l < lens[n]) val = __expf(w[idx] - m) / Z;
    w[idx] = val;
}

// ---------------------------------------------------------------------------
extern "C" void kernel_launch(void* const* d_in, const int* in_sizes, int n_in,
                              void* d_out, int out_size, void* d_ws, size_t ws_size,
                              hipStream_t stream) {
    (void)in_sizes; (void)n_in; (void)out_size; (void)ws_size;
    const float* H   = (const float*)d_in[0];   // (64,2048,512)
    const float* S   = (const float*)d_in[1];   // (64,512)
    const int*   len = (const int*)  d_in[2];   // (64,)
    const float* Wk  = (const float*)d_in[3];   // (512,128)
    // d_in[4] = bk: per-row constant shift in the energy -> cancels in softmax.
    const float* Wq  = (const float*)d_in[5];   // (512,128)
    const float* bq  = (const float*)d_in[6];   // (128,)
    const float* Wv  = (const float*)d_in[7];   // (512,128)
    const float* bv  = (const float*)d_in[8];   // (128,)

    float* ctx = (float*)d_out;                 // (64,128)
    float* w   = ctx + N_B * A_B;               // (64,2048)

    float* ws   = (float*)d_ws;
    float* q    = ws;                           // 64*128
    float* kq   = q   + N_B * A_B;              // 64*512
    float* ph   = kq  + N_B * D_B;              // 64*SPLIT*512
    float* pmz  = ph  + (size_t)N_B * SPLIT * D_B;   // 64*SPLIT*2
    float* hn   = pmz + N_B * SPLIT * 2;        // 64*512
    float* mzf  = hn  + N_B * D_B;              // 64*2

    // 1) q = S @ Wq + bq            (64x512 @ 512x128)
    wmma_gemm_f32<false><<<(N_B/16)*(A_B/16), 32, 0, stream>>>(
        S, Wq, bq, q, /*K=*/D_B, /*lda=*/D_B, /*ldb=*/A_B, /*ldd=*/A_B, A_B/16);
    // 2) kq = q @ Wk^T              (64x128 @ 128x512), B[k][n] = Wk[n*128+k]
    wmma_gemm_f32<true><<<(N_B/16)*(D_B/16), 32, 0, stream>>>(
        q, Wk, nullptr, kq, /*K=*/A_B, /*lda=*/A_B, /*ldb=*/A_B, /*ldd=*/D_B, D_B/16);
    // 3) single streaming pass over H (online softmax + weighted-sum of rows)
    attn_stream<<<N_B * SPLIT, 256, 0, stream>>>(H, kq, len, w, ph, pmz);
    // 4) combine splits -> normalized hbar, final (m, Z)
    combine_splits<<<N_B, 256, 0, stream>>>(ph, pmz, hn, mzf);
    // 5) energies -> masked softmax weights (in place in d_out)
    finalize_w<<<(N_B * L_B) / 256, 256, 0, stream>>>(w, mzf, len);
    // 6) context = hbar_norm @ Wv + bv   (64x512 @ 512x128)
    wmma_gemm_f32<false><<<(N_B/16)*(A_B/16), 32, 0, stream>>>(
        hn, Wv, bv, ctx, /*K=*/D_B, /*lda=*/D_B, /*ldb=*/A_B, /*ldd=*/A_B, A_B/16);
}